// ModelNew_25056839204979
// MI455X (gfx1250) — compile-verified
//
#include <hip/hip_runtime.h>

typedef __attribute__((ext_vector_type(2))) float v2f;
typedef __attribute__((ext_vector_type(8))) float v8f;

#define CIN   32
#define COUT  16
#define DIN   32
#define DOUT  63
#define PDIM  15
#define POOL_ELEMS (16 * COUT * PDIM * PDIM * PDIM)   // 864000
#define CHSTRIDE (DIN * DIN * DIN)                     // 32768 floats per (n,ci) volume

// ---------------------------------------------------------------------------
// Stage 0: zero pool accumulator + stats in workspace
// ---------------------------------------------------------------------------
__global__ void zero_ws_kernel(float* __restrict__ p, int nf) {
    int i = blockIdx.x * 256 + threadIdx.x;
    if (i < nf) p[i] = 0.0f;
}

// ---------------------------------------------------------------------------
// Stage 1: fused ConvTranspose3d (WMMA f32) + bias + BN-stat accumulation
//          + pooled-sum accumulation.
// Grid: (63, 16) = (od, n); block: 256 threads = 8 waves.
// Each wave computes a 16(Cout) x 16(ow, same parity) tile with
// V_WMMA_F32_16X16X4_F32; K = (valid parity taps) x Cin in chunks of 4.
//
// B-fragment loads are UNPREDICATED: an invalid tile column only corrupts
// its own D column (GEMM columns are independent), and the epilogue masks
// that column from stats/pooling. A min() address clamp (once per tap)
// prevents the single possible 1-element OOB read of x.
// ---------------------------------------------------------------------------
__global__ __launch_bounds__(256) void convt_bn_pool_stage1(
    const float* __restrict__ x,      // [16][32][32][32][32]
    const float* __restrict__ w,      // [32][16][3][3][3]
    const float* __restrict__ bias,   // [16]
    float* __restrict__ poolG,        // [16][16][15][15][15] (sum accumulator)
    float* __restrict__ statsG)       // [32]: sum[16], sumsq[16]
{
    __shared__ float wLDS[27 * COUT * CIN];   // 55296 B: wLDS[tap][co][ci]

    const int od  = blockIdx.x;
    const int n   = blockIdx.y;
    const int tid = threadIdx.x;

    // Preload weights, applying the reference's flip+transpose:
    // wt[co][ci][tap] = w[ci][co][26 - tap]  ->  wLDS[26-tap][co][ci] = w-flat
    for (int i = tid; i < 27 * COUT * CIN; i += 256) {
        const int ci  = i / (COUT * 27);
        const int r   = i % (COUT * 27);
        const int co  = r / 27;
        const int tap = r % 27;
        wLDS[((26 - tap) * COUT + co) * CIN + ci] = w[i];
    }
    __syncthreads();

    const int lane = tid & 31;
    const int wave = tid >> 5;
    const int half = lane >> 4;    // selects K pair (A/B) and channel+8 (C/D)
    const int col  = lane & 15;    // N index (spatial) / M index (A rows)

    const int pd  = od & 1;
    const int odp = od >> 1;
    const int kd0 = pd ? 0 : 1;
    const int nkd = pd ? 2 : 1;

    // Channel per accumulator row r is fixed: co = r + 8*half
    float biasr[8];
#pragma unroll
    for (int r = 0; r < 8; ++r) biasr[r] = bias[r + 8 * half];

    float sAcc[8], ssAcc[8];
#pragma unroll
    for (int r = 0; r < 8; ++r) { sAcc[r] = 0.0f; ssAcc[r] = 0.0f; }

    // Base of this n's input volume
    const float* xn = x + (size_t)n * CIN * CHSTRIDE;

    // 63 oh values x 4 w-tiles (pw=0: ow'=0..15,16..31 ; pw=1: ow'=0..15,16..30+mask)
    for (int tIdx = wave; tIdx < 63 * 4; tIdx += 8) {
        const int oh   = tIdx >> 2;
        const int wt   = tIdx & 3;
        const int pw   = wt >> 1;
        const int tw0  = (wt & 1) * 16;
        const int ph   = oh & 1;
        const int ohp  = oh >> 1;
        const int owmaxp    = pw ? 30 : 31;
        const bool colvalid = (tw0 + col) <= owmaxp;

        v8f acc = {};
        const int kh0 = ph ? 0 : 1;
        const int nkh = ph ? 2 : 1;
        const int kw0 = pw ? 0 : 1;
        const int nkw = pw ? 2 : 1;

        for (int ai = 0; ai < nkd; ++ai) {
            const int kd = kd0 + 2 * ai;
            const int id = odp + (pd ? (kd >> 1) : 0);
            for (int bi = 0; bi < nkh; ++bi) {
                const int kh = kh0 + 2 * bi;
                const int ih = ohp + (ph ? (kh >> 1) : 0);
                for (int wi = 0; wi < nkw; ++wi) {
                    const int kw  = kw0 + 2 * wi;
                    const int iw0 = tw0 + (pw ? (kw >> 1) : 0);
                    const int tap = kd * 9 + kh * 3 + kw;
                    const float* wrow = &wLDS[(tap * COUT + col) * CIN];
                    // Clamp: only the masked-out column can step 1 past the
                    // row end; clamp keeps the read in-bounds (value unused).
                    const int iwl = min(iw0 + col, DIN - 1);
                    // x[n][2*half][id][ih][iwl] base; ci advances via offsets
                    const float* px = xn
                        + ((size_t)(2 * half) * DIN * DIN * DIN)
                        + ((size_t)id * DIN + ih) * DIN + iwl;
                    __builtin_prefetch(px + 4 * CHSTRIDE, 0, 1);
#pragma unroll
                    for (int c4 = 0; c4 < CIN; c4 += 4) {
                        // A fragment (LDS, ds_load_b64): lane M=col, K=c4+2h,+1
                        v2f av = *(const v2f*)&wrow[c4 + 2 * half];
                        // B fragment: VGPR0 = x[ci=c4+2h][col],
                        //             VGPR1 = x[ci=c4+2h+1][col]
                        const float* pb = px + (size_t)c4 * CHSTRIDE;
                        v2f bv;
                        bv.x = pb[0];
                        bv.y = pb[CHSTRIDE];
                        acc = __builtin_amdgcn_wmma_f32_16x16x4_f32(
                            false, av, false, bv, (short)0, acc, false, false);
                    }
                }
            }
        }

        // Epilogue: bias, BN statistics, pooled-sum scatter.
        const int  ow     = 2 * (tw0 + col) + pw;
        const bool poolok = colvalid && (od < 60) && (oh < 60) && (ow < 60);

        float pv[8];
#pragma unroll
        for (int r = 0; r < 8; ++r) {
            const float v  = acc[r] + biasr[r];
            const float vv = colvalid ? v : 0.0f;
            sAcc[r]  += vv;
            ssAcc[r] += vv * vv;
            pv[r] = poolok ? v * (1.0f / 64.0f) : 0.0f;
        }

        // Columns (2k, 2k+1) share a pool bin (ow differs by 2, bin = ow>>2);
        // pair-sum so only even columns issue atomics. Validity is monotone
        // within a pair (odd poolable => even poolable), so zeros are safe.
#pragma unroll
        for (int r = 0; r < 8; ++r)
            pv[r] += __shfl_xor(pv[r], 1, 32);

        if (((col & 1) == 0) && poolok) {
            const size_t base =
                (((size_t)n * COUT + 8 * half) * PDIM + (od >> 2)) * (PDIM * PDIM)
                + (size_t)(oh >> 2) * PDIM + (ow >> 2);
#pragma unroll
            for (int r = 0; r < 8; ++r)
                atomicAdd(&poolG[base + (size_t)r * (PDIM * PDIM * PDIM)], pv[r]);
        }
    }

    // Per-channel stat reduction across the 16 columns of each half-wave.
#pragma unroll
    for (int r = 0; r < 8; ++r) {
        float s = sAcc[r], ss = ssAcc[r];
        for (int off = 1; off < 16; off <<= 1) {
            s  += __shfl_xor(s,  off, 32);
            ss += __shfl_xor(ss, off, 32);
        }
        if (col == 0) {
            const int co = r + 8 * half;
            atomicAdd(&statsG[co],      s);
            atomicAdd(&statsG[16 + co], ss);
        }
    }
}

// ---------------------------------------------------------------------------
// Stage 2: per-channel mean/var from accumulated stats, apply BN affine to
// pooled sums (pool is linear, so BN commutes with the pooled mean).
// ---------------------------------------------------------------------------
__global__ void finalize_kernel(const float* __restrict__ poolG,
                                const float* __restrict__ statsG,
                                const float* __restrict__ gamma,
                                const float* __restrict__ beta,
                                float* __restrict__ out, int nelem)
{
    int i = blockIdx.x * 256 + threadIdx.x;
    if (i >= nelem) return;
    const int c = (i / (PDIM * PDIM * PDIM)) % COUT;
    const float M    = 16.0f * 63.0f * 63.0f * 63.0f;
    const float mean = statsG[c] / M;
    const float var  = statsG[16 + c] / M - mean * mean;
    const float inv  = rsqrtf(var + 1e-5f);
    out[i] = (poolG[i] - mean) * inv * gamma[c] + beta[c];
}

// ---------------------------------------------------------------------------
extern "C" void kernel_launch(void* const* d_in, const int* in_sizes, int n_in,
                              void* d_out, int out_size, void* d_ws, size_t ws_size,
                              hipStream_t stream)
{
    const float* x     = (const float*)d_in[0];
    const float* w     = (const float*)d_in[1];
    const float* b     = (const float*)d_in[2];
    const float* gamma = (const float*)d_in[3];
    const float* beta  = (const float*)d_in[4];
    float* out = (float*)d_out;

    float* poolG  = (float*)d_ws;            // 864000 floats
    float* statsG = poolG + POOL_ELEMS;      // 32 floats (sum, sumsq)

    const int nzero = POOL_ELEMS + 32;
    zero_ws_kernel<<<(nzero + 255) / 256, 256, 0, stream>>>(poolG, nzero);

    dim3 grid1(DOUT, 16);   // (od, n)
    convt_bn_pool_stage1<<<grid1, 256, 0, stream>>>(x, w, b, poolG, statsG);

    finalize_kernel<<<(POOL_ELEMS + 255) / 256, 256, 0, stream>>>(
        poolG, statsG, gamma, beta, out, POOL_ELEMS);
}